// class_dsu_46806553592377
// MI455X (gfx1250) — compile-verified
//
#include <hip/hip_runtime.h>
#include <hip/hip_bf16.h>

typedef float v2f __attribute__((ext_vector_type(2)));
typedef float v8f __attribute__((ext_vector_type(8)));
typedef unsigned int u32x4 __attribute__((ext_vector_type(4)));
typedef int i32x4 __attribute__((ext_vector_type(4)));
typedef int i32x8 __attribute__((ext_vector_type(8)));

#define EPSV 1e-5f
#define BB 4
#define CC 64
#define HH 128
#define WW 256
#define HWW (HH*WW)            // 32768 pixels per image
#define HG 512
#define WG 1024
#define NCID 20                // 19 real classes + the never-occurring "255" slot
#define TILE 128               // pixels per staged tile
#define NT 4                   // tiles processed per block (acc stays in VGPRs)
#define PSTR 130               // LDS row stride in floats (even, conflict-free)
#define STATS_FLOATS (NCID*BB*CC)   // 5120
#define GTR_BYTES (BB*HWW)          // 131072 bytes of uint8 cids

#if defined(__has_builtin)
#  if __has_builtin(__builtin_amdgcn_tensor_load_to_lds) && \
      __has_builtin(__builtin_amdgcn_s_wait_tensorcnt)
#    define HAVE_TDM 1
#  else
#    define HAVE_TDM 0
#  endif
#else
#  define HAVE_TDM 0
#endif

// ---------------------------------------------------------------------------
// Pass 0: nearest-neighbor downsample gt (B,512,1024) -> uint8 cid (B,128,256)
//         and zero the sum/sumsq/count accumulators in workspace.
// ---------------------------------------------------------------------------
__global__ void __launch_bounds__(256)
k_prep(const int* __restrict__ gt, unsigned char* __restrict__ gtr,
       float* __restrict__ statsF) {
  int idx = blockIdx.x * 256 + threadIdx.x;       // 0 .. B*HW-1 = 131071
  int b   = idx >> 15;
  int pix = idx & (HWW - 1);
  int h   = pix >> 8;
  int w   = pix & (WW - 1);
  gtr[idx] = (unsigned char)gt[b * (HG * WG) + (h * 4) * WG + (w * 4)];
  if (idx < 2 * STATS_FLOATS + NCID * BB) statsF[idx] = 0.0f;   // sum, sq, cnt
}

#if HAVE_TDM
// Issue one TDM tile load: 64 rows (channels) x 128 fp32 (pixels), global row
// stride = HWW elements, into LDS with hardware padding of 2 DWORDs after
// every 128 DWORDs -> LDS row stride 130 floats (matches PSTR, kills bank
// conflicts for the WMMA B-fragment reads). All descriptor words are uniform.
__device__ __forceinline__ void tdm_issue_tile(const float* gaddr, const float* ldsp) {
  unsigned long long ga = (unsigned long long)(uintptr_t)gaddr;
  unsigned int laddr = (unsigned int)(uintptr_t)ldsp;   // LDS aperture: low 32 bits = LDS offset
  u32x4 g0;
  g0[0] = 1u;                                           // count=1, no gather
  g0[1] = laddr;                                        // lds_addr
  g0[2] = (unsigned int)ga;                             // global_addr[31:0]
  g0[3] = (unsigned int)(ga >> 32) | 0x80000000u;       // global_addr[56:32] | type=2
  i32x8 g1;
  // [17:16]=data_size(4B)=2, [20]=pad_enable, [24:22]=pad_interval(128DW)=6,
  // [31:25]=pad_amount(2DW)=1, workgroup_mask=0
  g1[0] = 0x03920000;
  g1[1] = (int)0x80000000;        // [47:32]=atomic_barrier=0, [63:48]=tensor_dim0.lo16 (32768)
  g1[2] = 0x00400000;             // [79:64]=tensor_dim0.hi16=0, [95:80]=tensor_dim1.lo16=64
  g1[3] = 0x00800000;             // [111:96]=tensor_dim1.hi16=0, [127:112]=tile_dim0=128
  g1[4] = 0x00000040;             // [143:128]=tile_dim1=64, [159:144]=tile_dim2=0
  g1[5] = 0x00008000;             // tensor_dim0_stride[31:0] = 32768
  g1[6] = 0;                      // stride0.hi16=0, stride1.lo16=0
  g1[7] = 0;
  i32x4 z4 = {0, 0, 0, 0};
  i32x8 z8 = {0, 0, 0, 0, 0, 0, 0, 0};
  __builtin_amdgcn_tensor_load_to_lds(g0, g1, z4, z4, z8, 0);
}
#endif

// ---------------------------------------------------------------------------
// Pass 1: segmented reduction via one-hot WMMA (V_WMMA_F32_16X16X4_F32):
//   S[cid,c] += onehot(cid,p)*x[p,c],  Q += onehot*x^2,  CNT += onehot*1
// Block = 256 threads (8 waves). Each block owns NT=4 tiles of 128 pixels of
// one image; tiles are staged by the Tensor Data Mover into a double-buffered
// LDS region and overlapped with WMMA compute (s_wait_tensorcnt pipelining).
// ---------------------------------------------------------------------------
__global__ void __launch_bounds__(256)
k_stats(const float* __restrict__ x, const unsigned char* __restrict__ gtr,
        float* __restrict__ sumB, float* __restrict__ sqB,
        float* __restrict__ cntB) {
  __shared__ float xs[2][CC * PSTR];   // double-buffered x tile [c][p]
  __shared__ float sumS[32 * CC];      // per-block class sums (rows 0..31)
  __shared__ float sqS[32 * CC];
  __shared__ float cntS[32];

  const int tid   = threadIdx.x;
  const int lane  = tid & 31;
  const int wid   = tid >> 5;
  const int blk   = blockIdx.x;
  const int b     = blk >> 6;                    // 64 blocks per image
  const int pblk0 = (blk & 63) * (TILE * NT);    // first pixel of this block

  // zero block-local accumulators (visible after the first barrier)
  for (int i = tid; i < 32 * CC; i += 256) { sumS[i] = 0.f; sqS[i] = 0.f; }
  if (tid < 32) cntS[tid] = 0.f;

  const float* xb = x + (size_t)b * CC * HWW;

#if HAVE_TDM
  if (wid == 0) {
    tdm_issue_tile(xb + pblk0, &xs[0][0]);
    __builtin_amdgcn_s_wait_tensorcnt(0);
  }
  __syncthreads();
#else
  for (int it = 0; it < 32; ++it) {
    int linear = it * 256 + tid;
    int c = linear >> 7, p = linear & (TILE - 1);
    xs[0][c * PSTR + p] = xb[(size_t)c * HWW + pblk0 + p];
  }
  __syncthreads();
#endif

  // Fragment geometry (wave32 16x16x4 f32):
  //  A (16x4): lane L<16 holds K={0,1}, L>=16 holds K={2,3}; row M = lane&15
  //  B (4x16): mirrored K split; col N = lane&15
  //  C (16x16): vgpr r -> M = r + (lane<16 ? 0 : 8); N = lane&15
  const int m  = lane & 15;
  const int k0 = (lane >> 4) * 2;

  const v8f zero8 = {0.f, 0.f, 0.f, 0.f, 0.f, 0.f, 0.f, 0.f};
  v8f accS[2][4], accQ[2][4], accC[2];
#pragma unroll
  for (int t = 0; t < 2; ++t) {
    accC[t] = zero8;
#pragma unroll
    for (int u = 0; u < 4; ++u) { accS[t][u] = zero8; accQ[t][u] = zero8; }
  }

  const unsigned short* gs = (const unsigned short*)(gtr + (size_t)b * HWW);
  const v2f bones = {1.f, 1.f};
  const int pwrel = 16 * wid;                    // this wave's pixels in tile

  for (int t = 0; t < NT; ++t) {
    const int pblk = pblk0 + t * TILE;
    const float* xt = &xs[t & 1][0];
#if HAVE_TDM
    if (wid == 0 && t + 1 < NT)                  // prefetch next tile (overlap)
      tdm_issue_tile(xb + pblk + TILE, &xs[(t + 1) & 1][0]);
#endif
#pragma unroll
    for (int s = 0; s < 4; ++s) {                // 4 K-steps of 4 pixels
      const int prel = pwrel + 4 * s;
      const unsigned short cc2 = gs[(pblk + prel + k0) >> 1];  // aligned u16
      const int c0 = cc2 & 255, c1 = cc2 >> 8;
      v2f a0, a1;
      a0.x = (c0 == m)      ? 1.f : 0.f;  a0.y = (c1 == m)      ? 1.f : 0.f;
      a1.x = (c0 == m + 16) ? 1.f : 0.f;  a1.y = (c1 == m + 16) ? 1.f : 0.f;
#pragma unroll
      for (int u = 0; u < 4; ++u) {
        const int ch = u * 16 + m;               // channel = column N
        float b0 = xt[ch * PSTR + prel + k0];
        float b1 = xt[ch * PSTR + prel + k0 + 1];
        v2f bf = {b0, b1};
        v2f bq = {b0 * b0, b1 * b1};
        accS[0][u] = __builtin_amdgcn_wmma_f32_16x16x4_f32(false, a0, false, bf, (short)0, accS[0][u], false, false);
        accS[1][u] = __builtin_amdgcn_wmma_f32_16x16x4_f32(false, a1, false, bf, (short)0, accS[1][u], false, false);
        accQ[0][u] = __builtin_amdgcn_wmma_f32_16x16x4_f32(false, a0, false, bq, (short)0, accQ[0][u], false, false);
        accQ[1][u] = __builtin_amdgcn_wmma_f32_16x16x4_f32(false, a1, false, bq, (short)0, accQ[1][u], false, false);
      }
      accC[0] = __builtin_amdgcn_wmma_f32_16x16x4_f32(false, a0, false, bones, (short)0, accC[0], false, false);
      accC[1] = __builtin_amdgcn_wmma_f32_16x16x4_f32(false, a1, false, bones, (short)0, accC[1], false, false);
    }
#if HAVE_TDM
    if (wid == 0 && t + 1 < NT)
      __builtin_amdgcn_s_wait_tensorcnt(0);      // next tile landed in LDS
    __syncthreads();
#else
    __syncthreads();                             // everyone done reading buf
    if (t + 1 < NT) {
      for (int it = 0; it < 32; ++it) {
        int linear = it * 256 + tid;
        int c = linear >> 7, p = linear & (TILE - 1);
        xs[(t + 1) & 1][c * PSTR + p] = xb[(size_t)c * HWW + pblk + TILE + p];
      }
      __syncthreads();
    }
#endif
  }

  // reduce the 8 waves' fragments through LDS
  const int rowAdd = (lane < 16) ? 0 : 8;
#pragma unroll
  for (int t = 0; t < 2; ++t) {
#pragma unroll
    for (int u = 0; u < 4; ++u) {
#pragma unroll
      for (int r = 0; r < 8; ++r) {
        int M = t * 16 + r + rowAdd;
        atomicAdd(&sumS[M * CC + u * 16 + m], accS[t][u][r]);
        atomicAdd(&sqS [M * CC + u * 16 + m], accQ[t][u][r]);
      }
    }
    if (m == 0) {                                // count replicated across N
#pragma unroll
      for (int r = 0; r < 8; ++r)
        atomicAdd(&cntS[t * 16 + r + rowAdd], accC[t][r]);
    }
  }
  __syncthreads();

  // flush block partials (rows 0..19 only) to global tables
  for (int i = tid; i < NCID * CC; i += 256) {
    int cid = i >> 6, c = i & 63;
    float vs = sumS[cid * CC + c];
    float vq = sqS[cid * CC + c];
    int o = (cid * BB + b) * CC + c;
    if (vs != 0.f) atomicAdd(&sumB[o], vs);
    if (vq != 0.f) atomicAdd(&sqB[o], vq);
  }
  if (tid < NCID) {
    float vc = cntS[tid];
    if (vc != 0.f) atomicAdd(&cntB[tid * BB + b], vc);
  }
}

// ---------------------------------------------------------------------------
// Pass 2: per-(cid,c) thread: mean/std per b, batch-variance (ddof=1) across
// B, then fold beta/gamma into per-(cid,b,c) scale/bias.
// ---------------------------------------------------------------------------
__global__ void k_params(const float* __restrict__ sumB, const float* __restrict__ sqB,
                         const float* __restrict__ cntB,
                         const float* __restrict__ eps_mu, const float* __restrict__ eps_std,
                         float* __restrict__ scaleB, float* __restrict__ biasB) {
  const int cid = blockIdx.x;      // 0..19
  const int c   = threadIdx.x;     // 0..63
  float mean[BB], stdv[BB];
#pragma unroll
  for (int b = 0; b < BB; ++b) {
    float cnt = cntB[cid * BB + b];
    float np  = (cnt > 0.5f) ? cnt : 1.0f;       // reference's npx==0 -> 1
    float s   = sumB[(cid * BB + b) * CC + c];
    float q   = sqB [(cid * BB + b) * CC + c];
    float mu  = s / np;
    float var = fmaxf(q / np - mu * mu, 0.0f);
    mean[b] = mu;
    stdv[b] = sqrtf(var + EPSV);
  }
  float mbar = 0.f, sbar = 0.f;
#pragma unroll
  for (int b = 0; b < BB; ++b) { mbar += mean[b]; sbar += stdv[b]; }
  mbar *= (1.0f / BB); sbar *= (1.0f / BB);
  float vm = 0.f, vs = 0.f;
#pragma unroll
  for (int b = 0; b < BB; ++b) {
    float dm = mean[b] - mbar, ds = stdv[b] - sbar;
    vm += dm * dm; vs += ds * ds;
  }
  vm *= (1.0f / (BB - 1)); vs *= (1.0f / (BB - 1));   // ddof=1
  const float sqm = sqrtf(vm + EPSV), sqs = sqrtf(vs + EPSV);
#pragma unroll
  for (int b = 0; b < BB; ++b) {
    int o = (cid * BB + b) * CC + c;
    float beta  = mean[b] + eps_mu[o]  * sqm;
    float gamma = stdv[b] + eps_std[o] * sqs;
    float sc = gamma / stdv[b];
    scaleB[o] = sc;
    biasB[o]  = beta - mean[b] * sc;
  }
}

// ---------------------------------------------------------------------------
// Pass 3: out = x * scale[cid] + bias[cid], float4-streamed (the bandwidth-
// critical pass: 32 MB in + 32 MB out). 20-entry LUT per (b,c) held in LDS.
// ---------------------------------------------------------------------------
__global__ void __launch_bounds__(256)
k_apply(const float* __restrict__ x, const unsigned char* __restrict__ gtr,
        const float* __restrict__ scaleB, const float* __restrict__ biasB,
        float* __restrict__ out) {
  __shared__ float sS[NCID], sB[NCID];
  const int blk = blockIdx.x;
  const int b   = blk >> 9;           // 512 blocks per image (64 c * 8 segs)
  const int c   = (blk >> 3) & 63;
  const int seg = blk & 7;
  if (threadIdx.x < NCID) {
    int o = (threadIdx.x * BB + b) * CC + c;
    sS[threadIdx.x] = scaleB[o];
    sB[threadIdx.x] = biasB[o];
  }
  __syncthreads();
  const size_t base = ((size_t)b * CC + c) * HWW;
  const unsigned int* gw = (const unsigned int*)(gtr + (size_t)b * HWW);
  const int p0 = seg * 4096;
#pragma unroll
  for (int it = 0; it < 4; ++it) {
    int p = p0 + it * 1024 + threadIdx.x * 4;
    float4 xv = *(const float4*)(x + base + p);
    unsigned int cw = gw[p >> 2];                 // 4 packed uint8 cids
    float4 ov;
    ov.x = fmaf(xv.x, sS[cw & 255],         sB[cw & 255]);
    ov.y = fmaf(xv.y, sS[(cw >> 8) & 255],  sB[(cw >> 8) & 255]);
    ov.z = fmaf(xv.z, sS[(cw >> 16) & 255], sB[(cw >> 16) & 255]);
    ov.w = fmaf(xv.w, sS[cw >> 24],         sB[cw >> 24]);
    *(float4*)(out + base + p) = ov;
  }
}

// ---------------------------------------------------------------------------
extern "C" void kernel_launch(void* const* d_in, const int* in_sizes, int n_in,
                              void* d_out, int out_size, void* d_ws, size_t ws_size,
                              hipStream_t stream) {
  const float* x       = (const float*)d_in[0];
  const int*   gt      = (const int*)d_in[1];
  const float* eps_mu  = (const float*)d_in[2];
  const float* eps_std = (const float*)d_in[3];
  float* out = (float*)d_out;

  // workspace layout: [uint8 gtr (128KB)] [sum][sq][cnt][scale][bias] floats
  unsigned char* gtr = (unsigned char*)d_ws;
  float* fbase  = (float*)((char*)d_ws + GTR_BYTES);
  float* sumB   = fbase;
  float* sqB    = fbase + STATS_FLOATS;
  float* cntB   = fbase + 2 * STATS_FLOATS;
  float* scaleB = cntB + NCID * BB;
  float* biasB  = scaleB + STATS_FLOATS;

  k_prep  <<<512,         256, 0, stream>>>(gt, gtr, fbase);
  k_stats <<<BB * 64,     256, 0, stream>>>(x, gtr, sumB, sqB, cntB);
  k_params<<<NCID,        CC,  0, stream>>>(sumB, sqB, cntB, eps_mu, eps_std, scaleB, biasB);
  k_apply <<<BB * CC * 8, 256, 0, stream>>>(x, gtr, scaleB, biasB, out);
}